// TGACN_8452495638653
// MI455X (gfx1250) — compile-verified
//
#include <hip/hip_runtime.h>
#include <math.h>

// ---------------- sizes ----------------
#define BB   32
#define CC   64
#define NN   512
#define TT   24
#define NT   (NN*TT)            // 12288
#define SZ   ((size_t)BB*CC*NT) // 25165824 floats per big tensor

typedef __attribute__((ext_vector_type(2))) float v2f;
typedef __attribute__((ext_vector_type(8))) float v8f;

__device__ __forceinline__ v8f wmma_f32(v2f a, v2f b, v8f c) {
  // V_WMMA_F32_16X16X4_F32 : D = A(16x4) * B(4x16) + C(16x16), fp32 throughout
  return __builtin_amdgcn_wmma_f32_16x16x4_f32(false, a, false, b, (short)0, c,
                                               false, false);
}

// ---------------- LayerNorm over T ----------------
__global__ __launch_bounds__(256)
void ln_kernel(const float* __restrict__ x, const float* __restrict__ w,
               const float* __restrict__ bvec, float* __restrict__ y) {
  size_t row = (size_t)blockIdx.x * blockDim.x + threadIdx.x; // (b,c,n)
  if (row >= (size_t)BB * CC * NN) return;
  const float* xp = x + row * TT;
  float* yp = y + row * TT;
  float v[TT];
  float s = 0.f;
#pragma unroll
  for (int t = 0; t < TT; ++t) { v[t] = xp[t]; s += v[t]; }
  float mu = s * (1.f / TT);
  float s2 = 0.f;
#pragma unroll
  for (int t = 0; t < TT; ++t) { float d = v[t] - mu; s2 += d * d; }
  float inv = rsqrtf(s2 * (1.f / TT) + 1e-5f);
#pragma unroll
  for (int t = 0; t < TT; ++t) yp[t] = (v[t] - mu) * inv * w[t] + bvec[t];
}

// ---------------- kc[b,c,t] = sum_n y[b,c,n,t]*alpha[n] ----------------
__global__ __launch_bounds__(256)
void kc_kernel(const float* __restrict__ y1, const float* __restrict__ alpha,
               float* __restrict__ kc) {
  int idx = blockIdx.x * 256 + threadIdx.x;         // (b*64+c)*24+t
  if (idx >= BB * CC * TT) return;
  int t = idx % TT;
  int bc = idx / TT;
  const float* p = y1 + (size_t)bc * NT + t;
  float s = 0.f;
  for (int n = 0; n < NN; ++n) s += p[(size_t)n * TT] * alpha[n];
  kc[idx] = s;
}

// ---------------- Ac = softmax(kc @ ca_W @ kc^T) per batch ----------------
__global__ __launch_bounds__(256)
void ac_kernel(const float* __restrict__ kc, const float* __restrict__ caW,
               float* __restrict__ Ac) {
  __shared__ float kcS[CC * TT];
  __shared__ float tmpS[CC * TT];
  __shared__ float S[CC * CC];
  int b = blockIdx.x, tid = threadIdx.x;
  for (int i = tid; i < CC * TT; i += 256) kcS[i] = kc[(size_t)b * CC * TT + i];
  __syncthreads();
  for (int i = tid; i < CC * TT; i += 256) {
    int c = i / TT, s = i % TT;
    float acc = 0.f;
    for (int t = 0; t < TT; ++t) acc += kcS[c * TT + t] * caW[t * TT + s];
    tmpS[i] = acc;
  }
  __syncthreads();
  for (int i = tid; i < CC * CC; i += 256) {
    int c = i >> 6, d = i & 63;
    float acc = 0.f;
    for (int s = 0; s < TT; ++s) acc += tmpS[c * TT + s] * kcS[d * TT + s];
    S[i] = acc;
  }
  __syncthreads();
  if (tid < CC) {
    float mx = -INFINITY;
    for (int d = 0; d < CC; ++d) mx = fmaxf(mx, S[tid * CC + d]);
    float sum = 0.f;
    for (int d = 0; d < CC; ++d) { float e = __expf(S[tid * CC + d] - mx); S[tid * CC + d] = e; sum += e; }
    float r = 1.f / sum;
    for (int d = 0; d < CC; ++d) Ac[(size_t)b * CC * CC + tid * CC + d] = S[tid * CC + d] * r;
  }
}

// ------------- generic 64x64 @ 64xNT batched WMMA GEMM (+bias/add/relu) -------------
// C[b] = A_b(64x64) @ B[b](64xNT) ; A_b = A + b*aStride (aStride==0 -> shared weights)
template <bool HAS_BIAS, bool HAS_ADD, bool DO_RELU>
__global__ __launch_bounds__(128)
void gemm64_wmma(const float* __restrict__ A, long long aStride,
                 const float* __restrict__ Bm, const float* __restrict__ addSrc,
                 const float* __restrict__ bias, float* __restrict__ Cm) {
  const int b = blockIdx.y;
  const int tid = threadIdx.x;
  const int lane = tid & 31;
  const int wid = tid >> 5;
  const int m0 = wid * 16;
  const int l15 = lane & 15;
  const int hi = lane >> 4;       // 0: K 0..1 / M rows 0..7 ; 1: K 2..3 / M rows 8..15
  const int koff = hi * 2;

  const float* Ab = A + (long long)b * aStride;
  const float* Bb = Bm + (size_t)b * CC * NT;
  float* Cb = Cm + (size_t)b * CC * NT;
  const float* Addb = HAS_ADD ? addSrc + (size_t)b * CC * NT : nullptr;

  v2f aF[16];
#pragma unroll
  for (int ks = 0; ks < 16; ++ks) {
    int k = ks * 4 + koff;
    const float* ap = Ab + (m0 + l15) * CC + k;
    aF[ks].x = ap[0];
    aF[ks].y = ap[1];
  }

  const int TPB = 6;  // 768 N-tiles / 128 blocks
  for (int tIdx = 0; tIdx < TPB; ++tIdx) {
    int n0 = (blockIdx.x * TPB + tIdx) * 16;
    v8f acc = {};
#pragma unroll
    for (int ks = 0; ks < 16; ++ks) {
      int k = ks * 4 + koff;
      const float* bp = Bb + (size_t)k * NT + n0 + l15;
      v2f bF;
      bF.x = bp[0];
      bF.y = bp[NT];
      acc = wmma_f32(aF[ks], bF, acc);
    }
    // branch-free specialized epilogue
    float addv[8];
    if (HAS_ADD) {
#pragma unroll
      for (int r = 0; r < 8; ++r)
        addv[r] = Addb[(size_t)(m0 + r + hi * 8) * NT + n0 + l15];
    }
#pragma unroll
    for (int r = 0; r < 8; ++r) {
      int row = m0 + r + hi * 8;
      size_t off = (size_t)row * NT + n0 + l15;
      float v = acc[r];
      if (HAS_BIAS) v += bias[row];
      if (HAS_ADD) v += addv[r];
      if (DO_RELU) v = fmaxf(v, 0.f);
      Cb[off] = v;
    }
  }
}

// ---------------- kg[b,n,t] = sum_c y2[b,c,n,t]*alpha[c] ----------------
__global__ __launch_bounds__(256)
void kg_kernel(const float* __restrict__ y2, const float* __restrict__ alpha,
               float* __restrict__ kg) {
  int idx = blockIdx.x * 256 + threadIdx.x;  // b*NN*TT + n*TT + t
  if (idx >= BB * NN * TT) return;
  int t = idx % TT;
  int n = (idx / TT) % NN;
  int b = idx / (NN * TT);
  const float* p = y2 + ((size_t)b * CC * NN + n) * TT + t;
  float s = 0.f;
  for (int c = 0; c < CC; ++c) s += p[(size_t)c * NT] * alpha[c];
  kg[idx] = s;
}

// ---------------- P[b,n,s] = sum_t kg[b,n,t]*ga_W[t,s] ----------------
__global__ __launch_bounds__(256)
void p_kernel(const float* __restrict__ kg, const float* __restrict__ gaW,
              float* __restrict__ P) {
  int idx = blockIdx.x * 256 + threadIdx.x;
  if (idx >= BB * NN * TT) return;
  int s = idx % TT;
  int bn = idx / TT;
  const float* kp = kg + (size_t)bn * TT;
  float acc = 0.f;
#pragma unroll
  for (int t = 0; t < TT; ++t) acc += kp[t] * gaW[t * TT + s];
  P[idx] = acc;
}

// ---------------- Ag[b,n,:] = softmax(P[b,n,:] . kg[b,m,:] + mask) ----------------
__global__ __launch_bounds__(256)
void ag_kernel(const float* __restrict__ P, const float* __restrict__ kg,
               const int* __restrict__ adj, float* __restrict__ Ag) {
  int n = blockIdx.x, b = blockIdx.y, tid = threadIdx.x;
  __shared__ float Prow[TT];
  __shared__ float red[256];
  if (tid < TT) Prow[tid] = P[((size_t)b * NN + n) * TT + tid];
  __syncthreads();
  float v[2];
#pragma unroll
  for (int j = 0; j < 2; ++j) {
    int m = tid + j * 256;
    const float* kp = kg + ((size_t)b * NN + m) * TT;
    float s = 0.f;
#pragma unroll
    for (int t = 0; t < TT; ++t) s += Prow[t] * kp[t];
    s += 9.0e9f * ((float)adj[(size_t)n * NN + m] - 1.f);
    v[j] = s;
  }
  red[tid] = fmaxf(v[0], v[1]);
  __syncthreads();
  for (int off = 128; off > 0; off >>= 1) {
    if (tid < off) red[tid] = fmaxf(red[tid], red[tid + off]);
    __syncthreads();
  }
  float mx = red[0];
  __syncthreads();
  float e0 = __expf(v[0] - mx), e1 = __expf(v[1] - mx);
  red[tid] = e0 + e1;
  __syncthreads();
  for (int off = 128; off > 0; off >>= 1) {
    if (tid < off) red[tid] += red[tid + off];
    __syncthreads();
  }
  float r = 1.f / red[0];
  float* op = Ag + ((size_t)b * NN + n) * NN;
  op[tid] = e0 * r;
  op[tid + 256] = e1 * r;
}

// ---------- y3[b,o] = Ag[b](512x512) @ w2[b,o](512x24) via WMMA (N=24 -> 16+8) ----------
// hi-half B loads are unconditional (stay inside workspace) + cndmask-zeroed:
// no EXEC manipulation in the 128-iteration inner loop.
__global__ __launch_bounds__(128)
void agmm_wmma(const float* __restrict__ Ag, const float* __restrict__ W2,
               float* __restrict__ Y3) {
  const int b = blockIdx.z;
  const int o = blockIdx.y;
  const int tid = threadIdx.x;
  const int lane = tid & 31;
  const int wid = tid >> 5;
  const int l15 = lane & 15;
  const int hi = lane >> 4;
  const int koff = hi * 2;
  const int m0 = (blockIdx.x * 4 + wid) * 16;

  const float* Ab = Ag + (size_t)b * NN * NN;
  const float* Bb = W2 + ((size_t)b * CC + o) * NT;
  float* Cb = Y3 + ((size_t)b * CC + o) * NT;
  const bool hiValid = (l15 < 8);  // columns 16..23 of the 24-wide B

  v8f accLo = {}, accHi = {};
  for (int ks = 0; ks < 128; ++ks) {
    int k = ks * 4 + koff;
    const float* ap = Ab + (size_t)(m0 + l15) * NN + k;
    v2f aF;
    aF.x = ap[0];
    aF.y = ap[1];
    const float* bp = Bb + (size_t)k * TT + l15;
    v2f bLo, bHi;
    bLo.x = bp[0];
    bLo.y = bp[TT];
    float h0 = bp[16];        // always load; value discarded for lanes 8..15
    float h1 = bp[16 + TT];
    bHi.x = hiValid ? h0 : 0.f;
    bHi.y = hiValid ? h1 : 0.f;
    accLo = wmma_f32(aF, bLo, accLo);
    accHi = wmma_f32(aF, bHi, accHi);
  }
#pragma unroll
  for (int r = 0; r < 8; ++r) {
    int row = m0 + r + hi * 8;
    float* cp = Cb + (size_t)row * TT;
    cp[l15] = accLo[r];
    if (hiValid) cp[16 + l15] = accHi[r];
  }
}

// ---------------- kt[b,t,n] = sum_c y3[b,c,n,t]*alpha[c] ----------------
__global__ __launch_bounds__(256)
void kt_kernel(const float* __restrict__ y3, const float* __restrict__ alpha,
               float* __restrict__ kt) {
  int idx = blockIdx.x * 256 + threadIdx.x;  // b*TT*NN + t*NN + n
  if (idx >= BB * TT * NN) return;
  int n = idx % NN;
  int t = (idx / NN) % TT;
  int b = idx / (NN * TT);
  const float* p = y3 + ((size_t)b * CC * NN + n) * TT + t;
  float s = 0.f;
  for (int c = 0; c < CC; ++c) s += p[(size_t)c * NT] * alpha[c];
  kt[idx] = s;
}

// ---------------- At = softmax((kt@W1^T)(kt@W2^T)^T) per batch ----------------
__global__ __launch_bounds__(256)
void at_kernel(const float* __restrict__ kt, const float* __restrict__ W1,
               const float* __restrict__ W2, float* __restrict__ At) {
  __shared__ float u1S[TT * 10];
  __shared__ float u2S[TT * 10];
  __shared__ float G[TT * TT];
  int b = blockIdx.x, tid = threadIdx.x;
  if (tid < TT * 10) {
    int t = tid / 10, j = tid % 10;
    const float* kp = kt + ((size_t)b * TT + t) * NN;
    float s1 = 0.f, s2 = 0.f;
    for (int n = 0; n < NN; ++n) {
      float kv = kp[n];
      s1 += kv * W1[(size_t)j * NN + n];
      s2 += kv * W2[(size_t)j * NN + n];
    }
    u1S[tid] = s1;
    u2S[tid] = s2;
  }
  __syncthreads();
  for (int i = tid; i < TT * TT; i += 256) {
    int t = i / TT, s = i % TT;
    float acc = 0.f;
#pragma unroll
    for (int j = 0; j < 10; ++j) acc += u1S[t * 10 + j] * u2S[s * 10 + j];
    G[i] = acc;
  }
  __syncthreads();
  if (tid < TT) {
    float mx = -INFINITY;
    for (int s = 0; s < TT; ++s) mx = fmaxf(mx, G[tid * TT + s]);
    float sum = 0.f;
    for (int s = 0; s < TT; ++s) { float e = __expf(G[tid * TT + s] - mx); G[tid * TT + s] = e; sum += e; }
    float r = 1.f / sum;
    for (int s = 0; s < TT; ++s) At[(size_t)b * TT * TT + tid * TT + s] = G[tid * TT + s] * r;
  }
}

// ---------------- y4[b,c,n,t] = sum_s At[b,t,s]*y3[b,c,n,s] ----------------
__global__ __launch_bounds__(256)
void applyat_kernel(const float* __restrict__ At, const float* __restrict__ y3,
                    float* __restrict__ y4) {
  int b = blockIdx.y;
  __shared__ float AtS[TT * TT];
  int tid = threadIdx.x;
  for (int i = tid; i < TT * TT; i += 256) AtS[i] = At[(size_t)b * TT * TT + i];
  __syncthreads();
  int idx = blockIdx.x * 256 + tid;  // < CC*NN*TT
  int t = idx % TT;
  int cn = idx / TT;
  const float* yp = y3 + ((size_t)b * CC * NN + cn) * TT;
  float s = 0.f;
#pragma unroll
  for (int ss = 0; ss < TT; ++ss) s += AtS[t * TT + ss] * yp[ss];
  y4[((size_t)b * CC * NN + cn) * TT + t] = s;
}

// -------- TCN layer: out[t] = bias + W1*in[t] + W0*in[t-d] (two WMMA GEMMs) --------
// Shifted-tap loads use a clamped (always-valid) address + cndmask zeroing:
// branch-free inner loop.
__global__ __launch_bounds__(128)
void tcn_wmma(const float* __restrict__ In, const float* __restrict__ W,
              const float* __restrict__ bias, float* __restrict__ Out, int d) {
  const int b = blockIdx.y;
  const int tid = threadIdx.x;
  const int lane = tid & 31;
  const int wid = tid >> 5;
  const int m0 = wid * 16;
  const int l15 = lane & 15;
  const int hi = lane >> 4;
  const int koff = hi * 2;

  v2f a0F[16], a1F[16];  // tap k=0 pairs with shifted input, tap k=1 with current
#pragma unroll
  for (int ks = 0; ks < 16; ++ks) {
    int k = ks * 4 + koff;
    const float* wp = W + ((size_t)(m0 + l15) * CC + k) * 2;
    a0F[ks].x = wp[0];
    a0F[ks].y = wp[2];
    a1F[ks].x = wp[1];
    a1F[ks].y = wp[3];
  }

  const float* Inb = In + (size_t)b * CC * NT;
  float* Outb = Out + (size_t)b * CC * NT;

  const int TPB = 6;
  for (int tIdx = 0; tIdx < TPB; ++tIdx) {
    int n0 = (blockIdx.x * TPB + tIdx) * 16;
    int col = n0 + l15;
    int t = col % TT;
    bool sh = (t >= d);
    int colS = col - d;
    if (colS < 0) colS = 0;  // clamped, always-valid address; value selected below
    v8f acc = {};
#pragma unroll
    for (int ks = 0; ks < 16; ++ks) {
      int k = ks * 4 + koff;
      const float* bp = Inb + (size_t)k * NT + col;
      const float* sp = Inb + (size_t)k * NT + colS;
      v2f bCur, bSh;
      bCur.x = bp[0];
      bCur.y = bp[NT];
      float u0 = sp[0];
      float u1 = sp[NT];
      bSh.x = sh ? u0 : 0.f;
      bSh.y = sh ? u1 : 0.f;
      acc = wmma_f32(a1F[ks], bCur, acc);
      acc = wmma_f32(a0F[ks], bSh, acc);
    }
#pragma unroll
    for (int r = 0; r < 8; ++r) {
      int row = m0 + r + hi * 8;
      Outb[(size_t)row * NT + col] = acc[r] + bias[row];
    }
  }
}

// ================================================================
extern "C" void kernel_launch(void* const* d_in, const int* in_sizes, int n_in,
                              void* d_out, int out_size, void* d_ws,
                              size_t ws_size, hipStream_t stream) {
  (void)in_sizes; (void)n_in; (void)out_size; (void)ws_size;
  const float* x       = (const float*)d_in[0];
  const int*   adj     = (const int*)d_in[1];
  const float* ln_w    = (const float*)d_in[2];
  const float* ln_b    = (const float*)d_in[3];
  const float* ca_W    = (const float*)d_in[4];
  const float* ca_alpha= (const float*)d_in[5];
  const float* ga_W    = (const float*)d_in[6];
  const float* ga_alpha= (const float*)d_in[7];
  const float* g_W     = (const float*)d_in[8];
  const float* ta_W1   = (const float*)d_in[9];
  const float* ta_W2   = (const float*)d_in[10];
  const float* ta_alpha= (const float*)d_in[11];
  const float* tcn_w   = (const float*)d_in[12];
  const float* tcn_b   = (const float*)d_in[13];
  const float* res_w   = (const float*)d_in[14];
  const float* res_b   = (const float*)d_in[15];
  float* out = (float*)d_out;

  float* ws = (float*)d_ws;
  float* Y1  = ws;                 // SZ
  float* Y2  = Y1 + SZ;            // SZ
  float* AgS = Y2 + SZ;            // B*N*N = 8388608
  float* kcS = AgS + (size_t)BB * NN * NN;   // 49152
  float* AcS = kcS + (size_t)BB * CC * TT;   // 131072
  float* kgS = AcS + (size_t)BB * CC * CC;   // 393216
  float* PS  = kgS + (size_t)BB * NN * TT;   // 393216
  float* ktS = PS  + (size_t)BB * NN * TT;   // 393216
  float* AtS = ktS + (size_t)BB * TT * NN;   // 18432

  // 1. LayerNorm : x -> Y1
  ln_kernel<<<4096, 256, 0, stream>>>(x, ln_w, ln_b, Y1);
  // 2. kc
  kc_kernel<<<192, 256, 0, stream>>>(Y1, ca_alpha, kcS);
  // 3. Ac softmax
  ac_kernel<<<32, 256, 0, stream>>>(kcS, ca_W, AcS);
  // 4. Y2 = Ac @ Y1   (WMMA, plain epilogue)
  gemm64_wmma<false, false, false><<<dim3(128, 32), 128, 0, stream>>>(
      AcS, (long long)CC * CC, Y1, nullptr, nullptr, Y2);
  // 5. kg
  kg_kernel<<<1536, 256, 0, stream>>>(Y2, ga_alpha, kgS);
  // 6. P, Ag softmax
  p_kernel<<<1536, 256, 0, stream>>>(kgS, ga_W, PS);
  ag_kernel<<<dim3(512, 32), 256, 0, stream>>>(PS, kgS, adj, AgS);
  // 7. Y1 = g_W @ Y2  (reordered: g_W commutes past Ag)  (WMMA, plain)
  gemm64_wmma<false, false, false><<<dim3(128, 32), 128, 0, stream>>>(
      g_W, 0, Y2, nullptr, nullptr, Y1);
  // 8. Y2 = Ag @ Y1 per (b,o)  (WMMA, 512x512x24)
  agmm_wmma<<<dim3(8, 64, 32), 128, 0, stream>>>(AgS, Y1, Y2);
  // 9. kt, At
  kt_kernel<<<1536, 256, 0, stream>>>(Y2, ta_alpha, ktS);
  at_kernel<<<32, 256, 0, stream>>>(ktS, ta_W1, ta_W2, AtS);
  // 10. Y1 = einsum(At, Y2)
  applyat_kernel<<<dim3(3072, 32), 256, 0, stream>>>(AtS, Y2, Y1);
  // 11-13. TCN layers d=1,2,4 (WMMA, ping-pong)
  tcn_wmma<<<dim3(128, 32), 128, 0, stream>>>(Y1, tcn_w + 0 * CC * CC * 2,
                                              tcn_b + 0 * CC, Y2, 1);
  tcn_wmma<<<dim3(128, 32), 128, 0, stream>>>(Y2, tcn_w + 1 * CC * CC * 2,
                                              tcn_b + 1 * CC, Y1, 2);
  tcn_wmma<<<dim3(128, 32), 128, 0, stream>>>(Y1, tcn_w + 2 * CC * CC * 2,
                                              tcn_b + 2 * CC, Y2, 4);
  // 14. out = relu(res_w @ x + res_b + Y2)  (WMMA, fused bias+add+relu epilogue)
  gemm64_wmma<true, true, true><<<dim3(128, 32), 128, 0, stream>>>(
      res_w, 0, x, Y2, res_b, out);
}